// SUNLayer_54013508715190
// MI455X (gfx1250) — compile-verified
//
#include <hip/hip_runtime.h>
#include <hip/hip_bf16.h>

// ---------------------------------------------------------------------------
// SUN layer for MI455X (gfx1250, wave32).
//   N=50000 nodes, E=800000 edges, C_in=C_out=128, all f32.
//   out = mask*(x@Wr^T+br) + (scatter_sum(x[dst]->src))@Wn^T + bn + gvec
//   gvec = mean(x,0)@Wg^T + bg
// GEMMs use V_WMMA_F32_16X16X4_F32 (full f32 — kernel is bandwidth bound,
// so no precision trade is warranted). One wave per 16x16 output tile.
// ---------------------------------------------------------------------------

typedef float v2f __attribute__((ext_vector_type(2)));
typedef float v8f __attribute__((ext_vector_type(8)));

#define C128 128

// ---- zero the workspace accumulators (local[N*128] + colsum[128]) ----------
__global__ __launch_bounds__(256) void sun_zero_kernel(float* p, long long n) {
    long long i = (long long)blockIdx.x * blockDim.x + threadIdx.x;
    if (i < n) p[i] = 0.0f;
}

// ---- edge scatter: local[src] += x[dst]  (one wave per edge) ---------------
__global__ __launch_bounds__(256) void sun_scatter_kernel(
    const float* __restrict__ x, const long long* __restrict__ edge_index,
    float* __restrict__ loc, int E)
{
    int wave = blockIdx.x * (blockDim.x >> 5) + (threadIdx.x >> 5);
    if (wave >= E) return;                      // wave-uniform exit
    int lane = threadIdx.x & 31;
    long long src = edge_index[wave];           // edge_index[0, e]
    long long dst = edge_index[(long long)E + wave];  // edge_index[1, e]
    const float4 v = *(const float4*)(x + dst * C128 + lane * 4);
    float* p = loc + src * C128 + lane * 4;
    atomicAdd(p + 0, v.x);
    atomicAdd(p + 1, v.y);
    atomicAdd(p + 2, v.z);
    atomicAdd(p + 3, v.w);
}

// ---- column sums of x (for the mean): thread c owns column c ---------------
__global__ __launch_bounds__(128) void sun_colsum_kernel(
    const float* __restrict__ x, float* __restrict__ colsum, int N)
{
    int c = threadIdx.x;                        // 0..127, coalesced across row
    float s = 0.0f;
    for (int r = blockIdx.x; r < N; r += gridDim.x)
        s += x[(long long)r * C128 + c];
    atomicAdd(&colsum[c], s);
}

// ---- gvec[n] = (colsum/N) . Wg[n,:] + bg[n]  (single 128-thread block) -----
__global__ __launch_bounds__(128) void sun_gvec_kernel(
    const float* __restrict__ colsum, const float* __restrict__ Wg,
    const float* __restrict__ bg, float* __restrict__ gvec, float invN)
{
    int n = threadIdx.x;
    float s = 0.0f;
#pragma unroll 8
    for (int k = 0; k < C128; ++k)
        s += colsum[k] * invN * Wg[n * C128 + k];
    gvec[n] = s + bg[n];
}

// ---- fused dual-GEMM epilogue: one wave computes one 16x16 tile of out -----
__global__ __launch_bounds__(256) void sun_gemm_kernel(
    const float* __restrict__ x, const float* __restrict__ loc,
    const unsigned char* __restrict__ mask,
    const float* __restrict__ Wr, const float* __restrict__ br_,
    const float* __restrict__ Wn, const float* __restrict__ bn_,
    const float* __restrict__ gvec, float* __restrict__ out, int N)
{
    const int wave = blockIdx.x * (blockDim.x >> 5) + (threadIdx.x >> 5);
    const int numTiles = (N / 16) * 8;          // N divisible by 16; 128/16 cols
    if (wave >= numTiles) return;               // wave-uniform -> EXEC all 1s

    const int lane = threadIdx.x & 31;
    const int rowTile = wave >> 3;              // which 16-row band
    const int colTile = wave & 7;               // which 16-col band
    const int m0 = rowTile * 16;
    const int n0 = colTile * 16;
    const int g = lane >> 4;                    // lane group 0/1
    const int l = lane & 15;
    const int rowA = m0 + l;                    // A-fragment row for this lane
    const int colB = n0 + l;                    // B-fragment (=W row) col

    const float* xrow = x   + (long long)rowA * C128;
    const float* lrow = loc + (long long)rowA * C128;
    const float* wr   = Wr  + (long long)colB * C128;   // B = W^T
    const float* wn   = Wn  + (long long)colB * C128;

    v8f cr = {};   // x @ Wr^T accumulator
    v8f cn = {};   // local @ Wn^T accumulator

#pragma unroll 8
    for (int kk = 0; kk < C128; kk += 4) {
        const int k = kk + 2 * g;               // per ISA 16x4 f32 A layout
        v2f a  = *(const v2f*)(xrow + k);
        v2f al = *(const v2f*)(lrow + k);
        v2f bR = *(const v2f*)(wr + k);
        v2f bN = *(const v2f*)(wn + k);
        cr = __builtin_amdgcn_wmma_f32_16x16x4_f32(false, a,  false, bR,
                                                   (short)0, cr, false, false);
        cn = __builtin_amdgcn_wmma_f32_16x16x4_f32(false, al, false, bN,
                                                   (short)0, cn, false, false);
    }

    const float brc = br_[colB];
    const float bnc = bn_[colB];
    const float gvc = gvec[colB];

#pragma unroll
    for (int j = 0; j < 8; ++j) {               // C/D: VGPR j -> row j + 8*g
        const int m = m0 + j + 8 * g;
        float v = cn[j] + bnc + gvc;            // non-root + global branches
        if (mask[m]) v += cr[j] + brc;          // root branch only where mask
        out[(long long)m * C128 + colB] = v;
    }
}

// ---------------------------------------------------------------------------
extern "C" void kernel_launch(void* const* d_in, const int* in_sizes, int n_in,
                              void* d_out, int out_size, void* d_ws, size_t ws_size,
                              hipStream_t stream)
{
    const float*         x    = (const float*)d_in[0];
    const long long*     ei   = (const long long*)d_in[1];   // int64 [2,E]
    const unsigned char* mask = (const unsigned char*)d_in[2]; // bool [N]
    const float*         Wr   = (const float*)d_in[3];
    const float*         br_  = (const float*)d_in[4];
    const float*         Wn   = (const float*)d_in[5];
    const float*         bn_  = (const float*)d_in[6];
    const float*         Wg   = (const float*)d_in[7];
    const float*         bg   = (const float*)d_in[8];
    float*               out  = (float*)d_out;

    const int N = in_sizes[0] / C128;   // 50000
    const int E = in_sizes[1] / 2;      // 800000

    // workspace: local[N*128] | colsum[128] | gvec[128]
    float* loc    = (float*)d_ws;
    float* colsum = loc + (long long)N * C128;
    float* gvec   = colsum + C128;

    // 1) zero accumulators (local + colsum)
    {
        long long nz = (long long)N * C128 + C128;
        int blocks = (int)((nz + 255) / 256);
        sun_zero_kernel<<<blocks, 256, 0, stream>>>(loc, nz);
    }

    // 2) edge scatter-add (one wave per edge, 8 waves per block)
    {
        int blocks = (E + 7) / 8;
        sun_scatter_kernel<<<blocks, 256, 0, stream>>>(x, ei, loc, E);
    }

    // 3) column sums for the mean
    sun_colsum_kernel<<<512, 128, 0, stream>>>(x, colsum, N);

    // 4) global branch GEMV
    sun_gvec_kernel<<<1, 128, 0, stream>>>(colsum, Wg, bg, gvec, 1.0f / (float)N);

    // 5) fused dual WMMA GEMM + mask/bias/global epilogue
    {
        int numTiles = (N / 16) * 8;            // 25000 waves
        int blocks = (numTiles + 7) / 8;        // 8 waves / block
        sun_gemm_kernel<<<blocks, 256, 0, stream>>>(x, loc, mask, Wr, br_,
                                                    Wn, bn_, gvec, out, N);
    }
}